// EncoderGCN2_75265006895439
// MI455X (gfx1250) — compile-verified
//
#include <hip/hip_runtime.h>

// GCN 2-layer, two branches.
// - Dense GEMMs: fp32 WMMA (V_WMMA_F32_16X16X4_F32), B (weights) register-resident
//   per wave, amortized over RT row tiles; A loaded as 8B vectors; ReLU fused
//   into layer-2 A loads.
// - Message passing: L2-resident float4 gathers + global_atomic_add_f32 scatter.

typedef __attribute__((ext_vector_type(2))) float v2f;
typedef __attribute__((ext_vector_type(8))) float v8f;

// ---------------------------------------------------------------- utilities
__global__ void fill_kernel(float* __restrict__ p, float v, int n) {
  int i = blockIdx.x * blockDim.x + threadIdx.x;
  if (i < n) p[i] = v;
}

__global__ void deg_kernel(const int* __restrict__ dst, float* __restrict__ deg, int E) {
  int e = blockIdx.x * blockDim.x + threadIdx.x;
  if (e < E) unsafeAtomicAdd(&deg[dst[e]], 1.0f);   // -> global_atomic_add_f32
}

__global__ void rsqrt_kernel(float* __restrict__ deg, int n) {
  int i = blockIdx.x * blockDim.x + threadIdx.x;
  if (i < n) deg[i] = rsqrtf(deg[i]);               // deg >= 1 (self-loop)
}

// ---------------------------------------------------------------- dense GEMM via WMMA
// H[N x ND] = X[N x K] @ W[K x ND], fp32, V_WMMA_F32_16X16X4_F32.
// grid = (Nrows/(16*RT), ND/64); block = 128 (4 waves, one 16-col tile each).
// B slice for the wave's col tile lives in registers (K/4 v2f = 64 VGPRs) and is
// reused for RT row tiles. Optional fused ReLU on the A operand.
template <int K, int ND, int RT, bool RELU_A>
__global__ void gemm_wmma_f32(const float* __restrict__ X,
                              const float* __restrict__ W,
                              float* __restrict__ H) {
  const int wave = threadIdx.x >> 5;
  const int lane = threadIdx.x & 31;
  const int col0 = blockIdx.y * 64 + wave * 16;

  const int m  = lane & 15;        // A: M row / B,D: N col
  const int kb = (lane >> 4) * 2;  // half-wave K-pair select

  // ---- preload B fragments (loop-invariant across row tiles)
  v2f b[K / 4];
#pragma unroll
  for (int kk = 0; kk < K / 4; ++kk) {
    b[kk].x = W[(size_t)(kk * 4 + kb) * ND + col0 + m];
    b[kk].y = W[(size_t)(kk * 4 + kb + 1) * ND + col0 + m];
  }

  int row0 = blockIdx.x * (16 * RT);
#pragma unroll 1
  for (int rt = 0; rt < RT; ++rt, row0 += 16) {
    // ---- A fragments: contiguous pair -> 8-byte vector load (global_load_b64)
    const float* __restrict__ xrow = X + (size_t)(row0 + m) * K + kb;
    v2f a[K / 4];
#pragma unroll
    for (int kk = 0; kk < K / 4; ++kk) {
      v2f av = *(const v2f*)(xrow + kk * 4);
      if (RELU_A) { av.x = fmaxf(av.x, 0.0f); av.y = fmaxf(av.y, 0.0f); }
      a[kk] = av;
    }

    // ---- 32 back-to-back WMMAs (accumulation chain on C)
    v8f c = {};
#pragma unroll
    for (int kk = 0; kk < K / 4; ++kk)
      c = __builtin_amdgcn_wmma_f32_16x16x4_f32(false, a[kk], false, b[kk],
                                                (short)0, c, false, false);

    // ---- store: VGPR r -> M=r (lanes 0-15) / M=r+8 (lanes 16-31), N=lane&15
    const int rbase = (lane >> 4) * 8;
#pragma unroll
    for (int r = 0; r < 8; ++r)
      H[(size_t)(row0 + rbase + r) * ND + col0 + m] = c[r];
  }
}

// ---------------------------------------------------------------- graph ops
// out[v][t] = bias[t] + h[v][t] * dis[v]^2   (self-loop + bias init), float4.
template <int D>
__global__ void selfloop_bias_kernel(const float* __restrict__ h,
                                     const float* __restrict__ dis,
                                     const float* __restrict__ bias,
                                     float* __restrict__ out, int n) {
  int i = blockIdx.x * blockDim.x + threadIdx.x;       // one thread = 4 features
  if (i < n * (D / 4)) {
    int v = i / (D / 4);
    int t = (i & (D / 4 - 1)) * 4;
    float w = dis[v];
    float ww = w * w;
    float4 hv = *(const float4*)(h + (size_t)v * D + t);
    float4 bv = *(const float4*)(bias + t);
    float4 o;
    o.x = bv.x + hv.x * ww; o.y = bv.y + hv.y * ww;
    o.z = bv.z + hv.z * ww; o.w = bv.w + hv.w * ww;
    *(float4*)(out + (size_t)v * D + t) = o;
  }
}

// out[dst] += h[src] * dis[src]*dis[dst] ; one thread = 4 features of one edge.
template <int D>
__global__ void scatter_kernel(const int* __restrict__ src,
                               const int* __restrict__ dst,
                               const float* __restrict__ h,
                               const float* __restrict__ dis,
                               float* __restrict__ out, int E) {
  int idx = blockIdx.x * blockDim.x + threadIdx.x;
  int e = idx / (D / 4);
  int t = (idx & (D / 4 - 1)) * 4;
  if (e < E) {
    int s = src[e], d = dst[e];
    float w = dis[s] * dis[d];
    float4 hv = *(const float4*)(h + (size_t)s * D + t);   // L2-resident gather
    float* o = out + (size_t)d * D + t;
    unsafeAtomicAdd(o + 0, hv.x * w);
    unsafeAtomicAdd(o + 1, hv.y * w);
    unsafeAtomicAdd(o + 2, hv.z * w);
    unsafeAtomicAdd(o + 3, hv.w * w);
  }
}

// ---------------------------------------------------------------- host side
static inline int cdiv(long long a, long long b) { return (int)((a + b - 1) / b); }

static void run_branch(const float* x, const int* ei, const float* W1,
                       const float* b1, const float* W2, const float* b2,
                       float* out_final, float* deg, float* h, float* out1,
                       int N, int E, hipStream_t stream) {
  const int IN = 128, HD = 128, OD = 64;
  const int RT = 5;                 // 3125 row tiles = 625 blocks * 5
  const int* src = ei;              // edge_index[0]
  const int* dst = ei + E;          // edge_index[1]

  // degrees -> dis = rsqrt(deg); deg starts at 1 (self loop)
  fill_kernel<<<cdiv(N, 256), 256, 0, stream>>>(deg, 1.0f, N);
  deg_kernel<<<cdiv(E, 256), 256, 0, stream>>>(dst, deg, E);
  rsqrt_kernel<<<cdiv(N, 256), 256, 0, stream>>>(deg, N);
  float* dis = deg;

  const int rowBlocks = N / (16 * RT);

  // ---- layer 1: h = x@W1 ; out1 = b1 + self + scatter   (ReLU deferred to GEMM2)
  gemm_wmma_f32<IN, HD, RT, false>
      <<<dim3(rowBlocks, HD / 64), 128, 0, stream>>>(x, W1, h);
  selfloop_bias_kernel<HD><<<cdiv((long long)N * HD / 4, 256), 256, 0, stream>>>(
      h, dis, b1, out1, N);
  scatter_kernel<HD><<<cdiv((long long)E * HD / 4, 256), 256, 0, stream>>>(
      src, dst, h, dis, out1, E);

  // ---- layer 2: h2 = relu(out1)@W2 (ReLU fused into A loads; reuse h buffer)
  gemm_wmma_f32<HD, OD, RT, true>
      <<<dim3(rowBlocks, 1), 128, 0, stream>>>(out1, W2, h);
  selfloop_bias_kernel<OD><<<cdiv((long long)N * OD / 4, 256), 256, 0, stream>>>(
      h, dis, b2, out_final, N);
  scatter_kernel<OD><<<cdiv((long long)E * OD / 4, 256), 256, 0, stream>>>(
      src, dst, h, dis, out_final, E);
}

extern "C" void kernel_launch(void* const* d_in, const int* in_sizes, int n_in,
                              void* d_out, int out_size, void* d_ws, size_t ws_size,
                              hipStream_t stream) {
  const float* x_feat = (const float*)d_in[0];
  const int*   x_ei   = (const int*)d_in[1];
  const float* y_feat = (const float*)d_in[2];
  const int*   y_ei   = (const int*)d_in[3];
  const float* W1x = (const float*)d_in[4];
  const float* b1x = (const float*)d_in[5];
  const float* W2x = (const float*)d_in[6];
  const float* b2x = (const float*)d_in[7];
  const float* W1y = (const float*)d_in[8];
  const float* b1y = (const float*)d_in[9];
  const float* W2y = (const float*)d_in[10];
  const float* b2y = (const float*)d_in[11];

  const int N = in_sizes[0] / 128;
  const int E = in_sizes[1] / 2;

  // workspace (branches run sequentially, buffers reused):
  //   deg/dis : N floats ; h : N*128 ; out1 : N*128
  float* deg  = (float*)d_ws;
  float* h    = deg + N;
  float* out1 = h + (size_t)N * 128;

  float* out_x = (float*)d_out;              // [N,64]
  float* out_y = out_x + (size_t)N * 64;     // [N,64]

  run_branch(x_feat, x_ei, W1x, b1x, W2x, b2x, out_x, deg, h, out1, N, E, stream);
  run_branch(y_feat, y_ei, W1y, b1y, W2y, b2y, out_y, deg, h, out1, N, E, stream);
}